// MHA_42563125903860
// MI455X (gfx1250) — compile-verified
//
#include <hip/hip_runtime.h>
#include <hip/hip_bf16.h>

#define B_SZ 32
#define H_SZ 32
#define S_SZ 2048
#define E_SZ 4096
#define D_SZ 128

typedef __attribute__((ext_vector_type(2))) float v2f;
typedef __attribute__((ext_vector_type(8))) float v8f;
typedef __attribute__((ext_vector_type(4))) unsigned u32x4;
typedef __attribute__((ext_vector_type(4))) int      i32x4;
typedef __attribute__((ext_vector_type(8))) int      i32x8;

#if __has_builtin(__builtin_amdgcn_tensor_load_to_lds) && \
    __has_builtin(__builtin_amdgcn_s_wait_tensorcnt)
#define HAVE_TDM 1
#endif

// ---------------------------------------------------------------------------
// Y = X @ W^T (+ bias), X:(32,K), W:(N,K) row-major, Y:(32,N).
// Block = 16 waves, one 32(M) x 16(N) tile per block, split-K across waves
// (each wave owns K/16 = 256), partials reduced through LDS. 256 blocks x 16
// waves = 4096 waves per GEMM -> enough loads in flight to saturate HBM.
// ---------------------------------------------------------------------------
__global__ __launch_bounds__(512) void gemm_xwT_wmma(
    const float* __restrict__ X, const float* __restrict__ W,
    const float* __restrict__ bias, float* __restrict__ Y,
    int N, int K)
{
  __shared__ float part[16][32][16];

  const int lane = threadIdx.x & 31;
  const int wave = threadIdx.x >> 5;          // 0..15, split-K index
  const int n0   = blockIdx.x * 16;           // N tile for whole block
  const int idx  = lane & 15;                 // M row (A) / N col (B)
  const int hi   = lane >> 4;                 // K sub-pair: lanes 16-31 = K 2,3
  const int kchunk = K >> 4;                  // 256
  const int kbase  = wave * kchunk;

  // ISA 32-bit A 16x4 layout: lane holds K = hi*2 + {0,1} of row idx.
  const float* xa0 = X + (size_t)idx * K        + kbase + hi * 2;  // M 0..15
  const float* xa1 = X + (size_t)(idx + 16) * K + kbase + hi * 2;  // M 16..31
  const float* wb  = W + (size_t)(n0 + idx) * K + kbase + hi * 2;  // B 4x16

  v8f c0 = {}; v8f c1 = {};
  for (int k = 0; k < kchunk; k += 64) {
    __builtin_prefetch(wb + k + 256, 0, 1);     // global_prefetch_b8, 1KB ahead
#pragma unroll
    for (int kk = 0; kk < 64; kk += 4) {
      v2f a0 = *(const v2f*)(xa0 + k + kk);
      v2f a1 = *(const v2f*)(xa1 + k + kk);
      v2f b  = *(const v2f*)(wb  + k + kk);
      c0 = __builtin_amdgcn_wmma_f32_16x16x4_f32(
               false, a0, false, b, (short)0, c0, false, false);
      c1 = __builtin_amdgcn_wmma_f32_16x16x4_f32(
               false, a1, false, b, (short)0, c1, false, false);
    }
  }

  // C/D 16x16 f32 layout: VGPR r, lane -> (M = r + hi*8, N = idx)
#pragma unroll
  for (int r = 0; r < 8; ++r) {
    part[wave][r + hi * 8][idx]      = c0[r];
    part[wave][r + hi * 8 + 16][idx] = c1[r];
  }
  __syncthreads();

  const int t   = threadIdx.x;                // 512 threads = 32x16 outputs
  const int row = t >> 4, col = t & 15;
  float acc = 0.0f;
#pragma unroll
  for (int w = 0; w < 16; ++w) acc += part[w][row][col];
  if (bias) acc += bias[n0 + col];
  Y[(size_t)row * N + n0 + col] = acc;
}

// ---------------------------------------------------------------------------
// Decode attention for one (b,h). 8 wave32s; lane l owns head-dims 4l..4l+3.
// q row staged into LDS by the Tensor Data Mover (guarded). Score/value loops
// unrolled x4 so 4 independent 512B key/value rows are in flight per wave.
// New-token K/V (position L-1) peeled out of the hot loops.
// ---------------------------------------------------------------------------
__global__ __launch_bounds__(256) void attn_decode(
    const float* __restrict__ Q,  const float* __restrict__ Kc,
    const float* __restrict__ Vc, const float* __restrict__ Kn,
    const float* __restrict__ Vn, float* __restrict__ O,
    const int* __restrict__ step_p)
{
  __shared__ float sc[S_SZ];
  __shared__ float red[16];
  __shared__ float accbuf[8][D_SZ];
  __shared__ float qs[D_SZ];

  const int b    = blockIdx.x / H_SZ;
  const int h    = blockIdx.x % H_SZ;
  const int tid  = threadIdx.x;
  const int lane = tid & 31;
  const int wave = tid >> 5;

  int L = *step_p + 1;
  if (L > S_SZ) L = S_SZ;
  const int Lc = L - 1;                         // keys taken from the cache

  const float scale = 0.015625f;                // 1/sqrt(4096)
  const float* qrow  = Q  + (size_t)b * E_SZ + h * D_SZ;
  const float* kbase = Kc + ((size_t)b * H_SZ + h) * (size_t)S_SZ * D_SZ;
  const float* vbase = Vc + ((size_t)b * H_SZ + h) * (size_t)S_SZ * D_SZ;
  const float* knrow = Kn + (size_t)b * E_SZ + h * D_SZ;
  const float* vnrow = Vn + (size_t)b * E_SZ + h * D_SZ;

  // ---- stage q into LDS (TDM when available) ----
  if (wave == 0) {
#ifdef HAVE_TDM
    const unsigned lds_off = (unsigned)(size_t)(&qs[0]);  // flat LDS low bits
    const unsigned long long ga = (unsigned long long)qrow;
    u32x4 g0;
    g0[0] = 1u;                                     // count=1, user mode
    g0[1] = lds_off;                                // lds_addr
    g0[2] = (unsigned)(ga & 0xFFFFFFFFu);           // global_addr[31:0]
    g0[3] = (unsigned)((ga >> 32) & 0x1FFFFFFu)     // global_addr[56:32]
            | (2u << 30);                           // type = 2 (image)
    i32x8 g1;
    g1[0] = 0x00020000;          // workgroup_mask=0, data_size=2 (4B)
    g1[1] = (int)(128u << 16);   // tensor_dim0 = 128
    g1[2] = (int)(1u << 16);     // tensor_dim1 = 1
    g1[3] = (int)(128u << 16);   // tile_dim0 = 128
    g1[4] = 1;                   // tile_dim1 = 1
    g1[5] = 128;                 // tensor_dim0_stride = 128
    g1[6] = 0;
    g1[7] = 0;
    i32x4 z4 = {0, 0, 0, 0};
#if __clang_major__ >= 23
    i32x8 z8 = {0, 0, 0, 0, 0, 0, 0, 0};
    __builtin_amdgcn_tensor_load_to_lds(g0, g1, z4, z4, z8, 0);
#else
    __builtin_amdgcn_tensor_load_to_lds(g0, g1, z4, z4, 0);
#endif
    __builtin_amdgcn_s_wait_tensorcnt(0);
#else
    *(float4*)&qs[lane * 4] = *(const float4*)(qrow + lane * 4);
#endif
  }
  __syncthreads();
  const float4 qv = *(const float4*)&qs[lane * 4];

  // ---- pass 1: scores + running max ----
  float m = -3.4e38f;
  int k = wave;
  for (; k + 24 < Lc; k += 32) {
    __builtin_prefetch(kbase + (size_t)(k + 32) * D_SZ + lane * 4, 0, 1);
    __builtin_prefetch(kbase + (size_t)(k + 56) * D_SZ + lane * 4, 0, 1);
    const float4 k0 = *(const float4*)(kbase + (size_t)(k)      * D_SZ + lane * 4);
    const float4 k1 = *(const float4*)(kbase + (size_t)(k +  8) * D_SZ + lane * 4);
    const float4 k2 = *(const float4*)(kbase + (size_t)(k + 16) * D_SZ + lane * 4);
    const float4 k3 = *(const float4*)(kbase + (size_t)(k + 24) * D_SZ + lane * 4);
    float p0 = qv.x*k0.x + qv.y*k0.y + qv.z*k0.z + qv.w*k0.w;
    float p1 = qv.x*k1.x + qv.y*k1.y + qv.z*k1.z + qv.w*k1.w;
    float p2 = qv.x*k2.x + qv.y*k2.y + qv.z*k2.z + qv.w*k2.w;
    float p3 = qv.x*k3.x + qv.y*k3.y + qv.z*k3.z + qv.w*k3.w;
#pragma unroll
    for (int off = 16; off > 0; off >>= 1) {
      p0 += __shfl_xor(p0, off, 32);
      p1 += __shfl_xor(p1, off, 32);
      p2 += __shfl_xor(p2, off, 32);
      p3 += __shfl_xor(p3, off, 32);
    }
    p0 *= scale; p1 *= scale; p2 *= scale; p3 *= scale;
    if (lane == 0) {
      sc[k] = p0; sc[k + 8] = p1; sc[k + 16] = p2; sc[k + 24] = p3;
    }
    m = fmaxf(m, fmaxf(fmaxf(p0, p1), fmaxf(p2, p3)));
  }
  for (; k < Lc; k += 8) {
    const float4 kv = *(const float4*)(kbase + (size_t)k * D_SZ + lane * 4);
    float p = qv.x*kv.x + qv.y*kv.y + qv.z*kv.z + qv.w*kv.w;
#pragma unroll
    for (int off = 16; off > 0; off >>= 1) p += __shfl_xor(p, off, 32);
    p *= scale;
    if (lane == 0) sc[k] = p;
    m = fmaxf(m, p);
  }
  if ((Lc & 7) == wave) {                       // peeled: the new token's key
    const float4 kv = *(const float4*)(knrow + lane * 4);
    float p = qv.x*kv.x + qv.y*kv.y + qv.z*kv.z + qv.w*kv.w;
#pragma unroll
    for (int off = 16; off > 0; off >>= 1) p += __shfl_xor(p, off, 32);
    p *= scale;
    if (lane == 0) sc[Lc] = p;
    m = fmaxf(m, p);
  }
  if (lane == 0) red[wave] = m;
  __syncthreads();
  float gm = red[0];
#pragma unroll
  for (int w = 1; w < 8; ++w) gm = fmaxf(gm, red[w]);

  // ---- softmax: exp + sum ----
  float ls = 0.0f;
  for (int kk = tid; kk < L; kk += 256) {
    const float e = __expf(sc[kk] - gm);
    sc[kk] = e;
    ls += e;
  }
#pragma unroll
  for (int off = 16; off > 0; off >>= 1) ls += __shfl_xor(ls, off, 32);
  __syncthreads();
  if (lane == 0) red[wave] = ls;
  __syncthreads();
  float gs = 0.0f;
#pragma unroll
  for (int w = 0; w < 8; ++w) gs += red[w];
  const float inv = 1.0f / gs;

  // ---- pass 2: weighted value accumulation ----
  float4 acc = make_float4(0.f, 0.f, 0.f, 0.f);
  k = wave;
  for (; k + 24 < Lc; k += 32) {
    __builtin_prefetch(vbase + (size_t)(k + 32) * D_SZ + lane * 4, 0, 1);
    __builtin_prefetch(vbase + (size_t)(k + 56) * D_SZ + lane * 4, 0, 1);
    const float4 v0 = *(const float4*)(vbase + (size_t)(k)      * D_SZ + lane * 4);
    const float4 v1 = *(const float4*)(vbase + (size_t)(k +  8) * D_SZ + lane * 4);
    const float4 v2 = *(const float4*)(vbase + (size_t)(k + 16) * D_SZ + lane * 4);
    const float4 v3 = *(const float4*)(vbase + (size_t)(k + 24) * D_SZ + lane * 4);
    const float w0 = sc[k], w1 = sc[k + 8], w2 = sc[k + 16], w3 = sc[k + 24];
    acc.x += w0*v0.x + w1*v1.x + w2*v2.x + w3*v3.x;
    acc.y += w0*v0.y + w1*v1.y + w2*v2.y + w3*v3.y;
    acc.z += w0*v0.z + w1*v1.z + w2*v2.z + w3*v3.z;
    acc.w += w0*v0.w + w1*v1.w + w2*v2.w + w3*v3.w;
  }
  for (; k < Lc; k += 8) {
    const float4 vv = *(const float4*)(vbase + (size_t)k * D_SZ + lane * 4);
    const float w = sc[k];
    acc.x += w*vv.x; acc.y += w*vv.y; acc.z += w*vv.z; acc.w += w*vv.w;
  }
  if ((Lc & 7) == wave) {                       // peeled: the new token's value
    const float4 vv = *(const float4*)(vnrow + lane * 4);
    const float w = sc[Lc];
    acc.x += w*vv.x; acc.y += w*vv.y; acc.z += w*vv.z; acc.w += w*vv.w;
  }
  *(float4*)&accbuf[wave][lane * 4] = acc;
  __syncthreads();
  if (tid < D_SZ) {
    float o = 0.0f;
#pragma unroll
    for (int w = 0; w < 8; ++w) o += accbuf[w][tid];
    O[(size_t)b * E_SZ + h * D_SZ + tid] = o * inv;
  }
}

// ---------------------------------------------------------------------------
extern "C" void kernel_launch(void* const* d_in, const int* in_sizes, int n_in,
                              void* d_out, int out_size, void* d_ws, size_t ws_size,
                              hipStream_t stream) {
  const float* x    = (const float*)d_in[0];
  const float* Wq   = (const float*)d_in[1];
  const float* Wk   = (const float*)d_in[2];
  const float* Wv   = (const float*)d_in[3];
  const float* Wo   = (const float*)d_in[4];
  const float* bo   = (const float*)d_in[5];
  const float* kc   = (const float*)d_in[6];
  const float* vc   = (const float*)d_in[7];
  const int*   step = (const int*)d_in[8];

  float* ws   = (float*)d_ws;
  float* q    = ws;                             // (32, 4096)
  float* kn   = ws + (size_t)1 * B_SZ * E_SZ;   // (32, 4096)
  float* vn   = ws + (size_t)2 * B_SZ * E_SZ;   // (32, 4096)
  float* attn = ws + (size_t)3 * B_SZ * E_SZ;   // (32, 4096)
  float* out  = (float*)d_out;                  // (32, 1, 4096)

  const dim3 blk(512);                 // 16 waves: split-K 16 on one N-tile
  const dim3 grd(E_SZ / 16);           // 256 N-tiles

  gemm_xwT_wmma<<<grd, blk, 0, stream>>>(x,    Wq, nullptr, q,   E_SZ, E_SZ);
  gemm_xwT_wmma<<<grd, blk, 0, stream>>>(x,    Wk, nullptr, kn,  E_SZ, E_SZ);
  gemm_xwT_wmma<<<grd, blk, 0, stream>>>(x,    Wv, nullptr, vn,  E_SZ, E_SZ);
  attn_decode  <<<dim3(B_SZ * H_SZ), dim3(256), 0, stream>>>(q, kc, vc, kn, vn, attn, step);
  gemm_xwT_wmma<<<grd, blk, 0, stream>>>(attn, Wo, bo,      out, E_SZ, E_SZ);
}